// ProposalLayer_39505109188894
// MI455X (gfx1250) — compile-verified
//
#include <hip/hip_runtime.h>
#include <hip/hip_bf16.h>

// ---------------------------------------------------------------------------
// ProposalLayer for MI455X (gfx1250):
//   K1 prep:    decode+clip boxes, build sortable u64 keys (score<<32 | ~idx)
//   K2 sort:    single-WGP bitonic sort of 16384 u64 keys entirely in LDS
//               (async DMA global->LDS in, async DMA LDS->global out)
//   K3 gather:  sorted boxes / sorted scores into workspace
//   K4 nms:     single-WGP greedy NMS, full sorted box table staged into
//               230KB of the 320KB WGP LDS via global_load_async_to_lds,
//               1 barrier per serial step, then writeout of blob + scores.
// ---------------------------------------------------------------------------

#define N_ANCHORS   14400
#define NPAD        16384
#define TOP_N       2000
#define NMS_THRESH  0.7f

#define PREP_THREADS 256
#define SORT_THREADS 1024
#define NMS_THREADS  1024

// ---- workspace layout (bytes) ----
#define WS_KEYS_OFF      0u                    // u64 x 16384 = 131072
#define WS_PROP_OFF      131072u               // float4 x 14400 = 230400
#define WS_SBOX_OFF      361472u               // float4 x 14400 = 230400
#define WS_SSCORE_OFF    591872u               // float  x 14400 = 57600

// ---- sort kernel dynamic-LDS ----
#define SORT_LDS_BYTES   (NPAD * 8u)           // 131072

// ---- NMS kernel dynamic-LDS layout (byte offsets within dynamic block) ----
#define LDS_SBOX_OFF     0u                    // float4 x 14400 = 230400
#define LDS_MASK_OFF     230400u               // u32 x 450 (= 1800, pad 1824)
#define LDS_MASK_WORDS   450
#define LDS_KEEP_OFF     232224u               // int x 2000 = 8000
#define LDS_TOTAL        240224u               // <= 320KB WGP LDS (CDNA5)

// ===========================================================================
// K1: bbox_transform_inv + clip_boxes + key build (pads get key 0)
// ===========================================================================
__global__ __launch_bounds__(PREP_THREADS)
void prep_kernel(const float* __restrict__ cls,      // (N,2)
                 const float* __restrict__ deltas,   // (N,4)
                 const float* __restrict__ anchors,  // (N,4)
                 const float* __restrict__ im_info,  // (3,)
                 float4* __restrict__ proposals,     // ws
                 unsigned long long* __restrict__ keys) {
    int i = blockIdx.x * blockDim.x + threadIdx.x;
    if (i >= NPAD) return;
    if (i >= N_ANCHORS) { keys[i] = 0ull; return; }

    float ax1 = anchors[i * 4 + 0];
    float ay1 = anchors[i * 4 + 1];
    float ax2 = anchors[i * 4 + 2];
    float ay2 = anchors[i * 4 + 3];
    float w  = ax2 - ax1 + 1.0f;
    float h  = ay2 - ay1 + 1.0f;
    float cx = ax1 + 0.5f * w;
    float cy = ay1 + 0.5f * h;

    float dx = deltas[i * 4 + 0];
    float dy = deltas[i * 4 + 1];
    float dw = deltas[i * 4 + 2];
    float dh = deltas[i * 4 + 3];

    float pcx = dx * w + cx;
    float pcy = dy * h + cy;
    float pw  = expf(dw) * w;
    float ph  = expf(dh) * h;

    float x1 = pcx - 0.5f * pw;
    float y1 = pcy - 0.5f * ph;
    float x2 = pcx + 0.5f * pw;
    float y2 = pcy + 0.5f * ph;

    float xmax = im_info[1] - 1.0f;   // width  -> x
    float ymax = im_info[0] - 1.0f;   // height -> y
    x1 = fminf(fmaxf(x1, 0.0f), xmax);
    y1 = fminf(fmaxf(y1, 0.0f), ymax);
    x2 = fminf(fmaxf(x2, 0.0f), xmax);
    y2 = fminf(fmaxf(y2, 0.0f), ymax);

    proposals[i] = make_float4(x1, y1, x2, y2);

    // score = cls_prob[:,1]; non-negative float bits are order-preserving.
    // low word = 0xFFFFFFFF - i so that descending u64 order reproduces
    // argsort(-scores)'s stable tie-break (smaller index first).
    unsigned sb = __float_as_uint(cls[i * 2 + 1]);
    keys[i] = ((unsigned long long)sb << 32) |
              (unsigned long long)(0xFFFFFFFFu - (unsigned)i);
}

// ===========================================================================
// K2: single-WGP bitonic sort, descending, entirely in LDS (128KB of the
// 320KB WGP LDS). Keys come in and go out via the CDNA5 async DMA engine.
// ===========================================================================
__global__ __launch_bounds__(SORT_THREADS)
void bitonic_sort_kernel(unsigned long long* __restrict__ keys) {
    extern __shared__ char smem[];
    unsigned long long* lk = (unsigned long long*)smem;
    const unsigned tid = threadIdx.x;
    const unsigned lds_base = __builtin_amdgcn_groupstaticsize();

    // ---- async DMA global -> LDS: 16384 u64 as 8192 B128 chunks ----
    for (unsigned c = tid; c < NPAD / 2u; c += SORT_THREADS) {
        unsigned lds_off = lds_base + c * 16u;
        const unsigned long long* gp = keys + c * 2u;
        asm volatile("global_load_async_to_lds_b128 %0, %1, off"
                     :: "v"(lds_off), "v"(gp) : "memory");
    }
    asm volatile("s_wait_asynccnt 0" ::: "memory");
    __syncthreads();

    // ---- bitonic sort in LDS (105 barrier-separated passes) ----
    for (unsigned k = 2; k <= NPAD; k <<= 1) {
        for (unsigned j = k >> 1; j > 0; j >>= 1) {
            for (unsigned t = tid; t < NPAD; t += SORT_THREADS) {
                unsigned ixj = t ^ j;
                if (ixj > t) {
                    unsigned long long a = lk[t];
                    unsigned long long b = lk[ixj];
                    bool up = ((t & k) == 0);            // descending region
                    bool sw = up ? (a < b) : (a > b);
                    if (sw) { lk[t] = b; lk[ixj] = a; }
                }
            }
            __syncthreads();
        }
    }

    // ---- async DMA LDS -> global writeback ----
    for (unsigned c = tid; c < NPAD / 2u; c += SORT_THREADS) {
        unsigned lds_off = lds_base + c * 16u;
        unsigned long long* gp = keys + c * 2u;
        asm volatile("global_store_async_from_lds_b128 %0, %1, off"
                     :: "v"(gp), "v"(lds_off) : "memory");
    }
    asm volatile("s_wait_asynccnt 0" ::: "memory");
}

// ===========================================================================
// K3: gather sorted boxes + scores from sorted keys
// ===========================================================================
__global__ __launch_bounds__(PREP_THREADS)
void gather_kernel(const unsigned long long* __restrict__ keys,
                   const float4* __restrict__ proposals,
                   float4* __restrict__ sboxes,
                   float* __restrict__ sscores) {
    int i = blockIdx.x * blockDim.x + threadIdx.x;
    if (i >= N_ANCHORS) return;
    unsigned long long k = keys[i];
    unsigned idx = 0xFFFFFFFFu - (unsigned)(k & 0xFFFFFFFFull);
    sboxes[i]  = proposals[idx];
    sscores[i] = __uint_as_float((unsigned)(k >> 32));
}

// ===========================================================================
// K4: greedy NMS in one workgroup. Full sorted box table (230KB) is DMA'd
// into LDS with global_load_async_to_lds_b128 (ASYNCcnt path), then the
// serial suppression loop runs entirely out of LDS with ONE barrier per
// step: every thread redundantly computes the take/skip decision from the
// LDS mask bit and keeps an identical private kept-count. Writes d_out.
// ===========================================================================
__global__ __launch_bounds__(NMS_THREADS)
void nms_kernel(const float4* __restrict__ sboxes_g,
                const float*  __restrict__ sscores_g,
                float* __restrict__ out) {
    extern __shared__ char smem[];
    float4*   sb   = (float4*)(smem + LDS_SBOX_OFF);
    unsigned* mask = (unsigned*)(smem + LDS_MASK_OFF);
    int*      keep = (int*)(smem + LDS_KEEP_OFF);
    const int tid  = threadIdx.x;
    const unsigned lds_base = __builtin_amdgcn_groupstaticsize();

    // init suppression mask
    for (int w = tid; w < LDS_MASK_WORDS; w += NMS_THREADS) mask[w] = 0u;

    // ---- async DMA global -> LDS of the sorted box table (per-lane B128) ----
    for (unsigned idx = (unsigned)tid; idx < (unsigned)N_ANCHORS;
         idx += NMS_THREADS) {
        unsigned lds_off = lds_base + LDS_SBOX_OFF + idx * 16u;
        const float4* gp = sboxes_g + idx;
        asm volatile("global_load_async_to_lds_b128 %0, %1, off"
                     :: "v"(lds_off), "v"(gp) : "memory");
    }
    asm volatile("s_wait_asynccnt 0" ::: "memory");
    __syncthreads();

    // ---- greedy sequential NMS: one barrier per step ----
    int cnt = 0;                         // identical in every thread
    for (int i = 0; i < N_ANCHORS; ++i) {
        unsigned sup = (mask[i >> 5] >> (i & 31)) & 1u;   // LDS broadcast read
        if (!sup) {
            if (tid == 0) keep[cnt] = i;
            cnt++;
            float4 bi = sb[i];
            float  ai = (bi.z - bi.x) * (bi.w - bi.y);
            for (int j = i + 1 + tid; j < N_ANCHORS; j += NMS_THREADS) {
                unsigned mw = mask[j >> 5];
                if (!((mw >> (j & 31)) & 1u)) {
                    float4 bj = sb[j];
                    float ix1 = fmaxf(bi.x, bj.x);
                    float iy1 = fmaxf(bi.y, bj.y);
                    float ix2 = fminf(bi.z, bj.z);
                    float iy2 = fminf(bi.w, bj.w);
                    float iw = fmaxf(ix2 - ix1, 0.0f);
                    float ih = fmaxf(iy2 - iy1, 0.0f);
                    float inter = iw * ih;
                    float aj = (bj.z - bj.x) * (bj.w - bj.y);
                    float iou = inter / (ai + aj - inter + 1e-10f);
                    if (iou > NMS_THRESH)
                        atomicOr(&mask[j >> 5], 1u << (j & 31));
                }
            }
        }
        __syncthreads();                 // publish this step's suppressions
        if (cnt >= TOP_N) break;         // uniform across all threads
    }

    // ---- writeout: blob (2000x5, col0 = 0) then scores (2000) ----
    float* blob   = out;              // 2000 * 5
    float* sc_out = out + TOP_N * 5;  // 2000
    for (int t = tid; t < TOP_N; t += NMS_THREADS) {
        float x1 = 0.f, y1 = 0.f, x2 = 0.f, y2 = 0.f, s = 0.f;
        if (t < cnt) {
            int p = keep[t];
            float4 b = sb[p];
            x1 = b.x; y1 = b.y; x2 = b.z; y2 = b.w;
            s = sscores_g[p];
        }
        blob[t * 5 + 0] = 0.0f;
        blob[t * 5 + 1] = x1;
        blob[t * 5 + 2] = y1;
        blob[t * 5 + 3] = x2;
        blob[t * 5 + 4] = y2;
        sc_out[t] = s;
    }
}

// ===========================================================================
extern "C" void kernel_launch(void* const* d_in, const int* in_sizes, int n_in,
                              void* d_out, int out_size, void* d_ws, size_t ws_size,
                              hipStream_t stream) {
    const float* cls     = (const float*)d_in[0];  // rpn_cls_prob (14400,2)
    const float* deltas  = (const float*)d_in[1];  // rpn_bbox_pred (14400,4)
    const float* anchors = (const float*)d_in[2];  // anchors (14400,4)
    const float* im_info = (const float*)d_in[3];  // (3,)
    float* out = (float*)d_out;                    // 2000*5 + 2000 floats

    char* ws = (char*)d_ws;
    unsigned long long* keys = (unsigned long long*)(ws + WS_KEYS_OFF);
    float4* proposals        = (float4*)(ws + WS_PROP_OFF);
    float4* sboxes           = (float4*)(ws + WS_SBOX_OFF);
    float*  sscores          = (float*)(ws + WS_SSCORE_OFF);

    prep_kernel<<<NPAD / PREP_THREADS, PREP_THREADS, 0, stream>>>(
        cls, deltas, anchors, im_info, proposals, keys);

    bitonic_sort_kernel<<<1, SORT_THREADS, SORT_LDS_BYTES, stream>>>(keys);

    gather_kernel<<<(N_ANCHORS + PREP_THREADS - 1) / PREP_THREADS,
                    PREP_THREADS, 0, stream>>>(keys, proposals, sboxes, sscores);

    nms_kernel<<<1, NMS_THREADS, LDS_TOTAL, stream>>>(sboxes, sscores, out);
}